// GlobalPointer_15530601742883
// MI455X (gfx1250) — compile-verified
//
#include <hip/hip_runtime.h>

typedef __attribute__((ext_vector_type(2))) float v2f;
typedef __attribute__((ext_vector_type(8))) float v8f;

#define HIDDEN_ 768
#define NOUT_   1536
#define BATCH_  4
#define SEQ_    1024
#define ENT_    12
#define DIMD_   64
#define NEGV_   1000000000000.0f

// ---------------------------------------------------------------------------
// Kernel 1: out = lhs @ W + bias, fused RoPE, scatter to Q/K buffers.
// Wave computes a 16(M) x 64(N) strip: 4 accumulators sharing one A fragment.
// ---------------------------------------------------------------------------
__global__ __launch_bounds__(256) void gp_qk_gemm_rope(
    const float* __restrict__ lhs,   // [4096][768]
    const float* __restrict__ W,     // [768][1536]
    const float* __restrict__ bias,  // [1536]
    float* __restrict__ Qbuf,        // [48][1024][64]
    float* __restrict__ Kbuf)        // [48][1024][64]
{
    const int lane = threadIdx.x & 31;
    const int wave = threadIdx.x >> 5;
    const int job  = blockIdx.x * 8 + wave;        // 6144 wave-jobs total
    const int mt   = job / (NOUT_ / 64);           // 0..255  (M tile of 16)
    const int nb   = (job % (NOUT_ / 64)) * 64;    // N base (strip of 64)
    const int lo   = lane & 15;
    const int hi   = lane >> 4;
    const int m    = mt * 16 + lo;                 // A-fragment row

    v8f acc[4];
#pragma unroll
    for (int j = 0; j < 4; ++j) {
        float bv = bias[nb + j * 16 + lo];         // bias per column, all rows
        v8f t = {bv, bv, bv, bv, bv, bv, bv, bv};
        acc[j] = t;
    }

    const float* arow = lhs + (size_t)m * HIDDEN_;
    for (int k0 = 0; k0 < HIDDEN_; k0 += 4) {
        const int kb = k0 + 2 * hi;
        v2f a = *(const v2f*)(arow + kb);          // A[m][kb], A[m][kb+1]
#pragma unroll
        for (int j = 0; j < 4; ++j) {
            const int n = nb + j * 16 + lo;
            v2f b;
            b.x = W[(size_t)kb * NOUT_ + n];       // B[kb][n]
            b.y = W[(size_t)(kb + 1) * NOUT_ + n]; // B[kb+1][n]
            acc[j] = __builtin_amdgcn_wmma_f32_16x16x4_f32(
                false, a, false, b, (short)0, acc[j], false, false);
        }
    }

    // RoPE: pair partner (dim^1) lives in lane^1, same VGPR index r.
#pragma unroll
    for (int j = 0; j < 4; ++j) {
        const int n    = nb + j * 16 + lo;
        const int h    = n >> 7;                   // head
        const int dcol = n & 127;                  // 0..127 within head
        const int dim  = dcol & 63;                // 0..63 within q or k
        const float sgn = (dim & 1) ? 1.0f : -1.0f;
        const float inv = __powf(10000.0f, -(float)(dim & ~1) * (1.0f / 64.0f));
        float* dst = (dcol < DIMD_) ? Qbuf : Kbuf;
#pragma unroll
        for (int r = 0; r < 8; ++r) {
            const int mm   = mt * 16 + r + 8 * hi; // row of this C element
            const int pos  = mm & (SEQ_ - 1);
            const int bidx = mm >> 10;
            float val  = acc[j][r];
            float part = __shfl_xor(val, 1);       // pair value from lane^1
            float ang  = (float)pos * inv;
            float s = __sinf(ang), c = __cosf(ang);
            float rot = val * c + sgn * part * s;
            dst[(((size_t)bidx * ENT_ + h) * SEQ_ + pos) * DIMD_ + dim] = rot;
        }
    }
}

// ---------------------------------------------------------------------------
// Kernel 2: logits[bh][m][n] = (Q[bh] @ K[bh]^T masked) / 8
// Block: 64x64 tile of one (b,h); Q/K tiles staged in LDS (stride 68 pad).
// ---------------------------------------------------------------------------
__global__ __launch_bounds__(256) void gp_logits(
    const float* __restrict__ Qbuf,
    const float* __restrict__ Kbuf,
    const int* __restrict__ amask,   // [4][1024]
    float* __restrict__ out)         // [48][1024][1024]
{
    __shared__ float qs[64 * 68];
    __shared__ float ks[64 * 68];

    const int bh   = blockIdx.z;
    const int bidx = bh / ENT_;
    const int m0   = blockIdx.y * 64;
    const int n0   = blockIdx.x * 64;
    const int tid  = threadIdx.x;

    // Stage 64x64 Q and K tiles (coalesced along dim)
    for (int idx = tid; idx < 64 * 64; idx += 256) {
        const int row = idx >> 6, col = idx & 63;
        qs[row * 68 + col] = Qbuf[((size_t)bh * SEQ_ + m0 + row) * DIMD_ + col];
        ks[row * 68 + col] = Kbuf[((size_t)bh * SEQ_ + n0 + row) * DIMD_ + col];
    }
    __syncthreads();

    const int lane = tid & 31;
    const int wave = tid >> 5;
    const int lo   = lane & 15;
    const int hi   = lane >> 4;

#pragma unroll
    for (int t = 0; t < 2; ++t) {
        const int tile = wave + t * 8;             // 16 tiles / 8 waves
        const int mt = tile >> 2, nt = tile & 3;
        v8f acc = {0.f, 0.f, 0.f, 0.f, 0.f, 0.f, 0.f, 0.f};
        const float* qrow = qs + (mt * 16 + lo) * 68;
        const float* krow = ks + (nt * 16 + lo) * 68;
#pragma unroll
        for (int k0 = 0; k0 < DIMD_; k0 += 4) {
            v2f a = *(const v2f*)(qrow + k0 + 2 * hi);
            v2f b = *(const v2f*)(krow + k0 + 2 * hi);
            acc = __builtin_amdgcn_wmma_f32_16x16x4_f32(
                false, a, false, b, (short)0, acc, false, false);
        }
        const int n  = n0 + nt * 16 + lo;
        const float pm = (float)amask[bidx * SEQ_ + n];
#pragma unroll
        for (int r = 0; r < 8; ++r) {
            const int m = m0 + mt * 16 + r + 8 * hi;
            float val = acc[r];
            val = val * pm - (1.0f - pm) * NEGV_;  // padding mask
            if (m > n) val -= NEGV_;               // strict lower-tri mask
            val *= 0.125f;                         // / sqrt(64)
            out[((size_t)bh * SEQ_ + m) * SEQ_ + n] = val;
        }
    }
}

// ---------------------------------------------------------------------------
extern "C" void kernel_launch(void* const* d_in, const int* in_sizes, int n_in,
                              void* d_out, int out_size, void* d_ws, size_t ws_size,
                              hipStream_t stream) {
    const float* lhs   = (const float*)d_in[0];  // (4,1024,768) f32
    const int*   amask = (const int*)d_in[1];    // (4,1024) i32
    const float* W     = (const float*)d_in[2];  // (768,1536) f32
    const float* bias  = (const float*)d_in[3];  // (1536,) f32
    float* out = (float*)d_out;                  // (4,12,1024,1024) f32

    float* Qbuf = (float*)d_ws;                              // 12 MB
    float* Kbuf = Qbuf + (size_t)BATCH_ * ENT_ * SEQ_ * DIMD_; // 12 MB

    // 6144 wave-jobs (256 M-tiles x 24 N-strips), 8 waves/block
    gp_qk_gemm_rope<<<768, 256, 0, stream>>>(lhs, W, bias, Qbuf, Kbuf);

    // 16x16 grid of 64x64 tiles per (b,h); 48 (b,h) pairs
    gp_logits<<<dim3(16, 16, 48), 256, 0, stream>>>(Qbuf, Kbuf, amask, out);
}